// MGA_50766513439346
// MI455X (gfx1250) — compile-verified
//
#include <hip/hip_runtime.h>

#define EMB 64
#define TILE_ROWS 16
#define ROWSTRIDE 68            // padded row stride (floats) -> conflict-free ds_load_b64
#define WAVES_PER_BLOCK 8
#define BLOCK_THREADS (WAVES_PER_BLOCK * 32)

typedef float v2f __attribute__((ext_vector_type(2)));
typedef float v8f __attribute__((ext_vector_type(8)));

// broadcast value held on lane `srcLane` (0..15) to the calling lane
__device__ __forceinline__ float lane_bcast(float v, int srcLane) {
    return __int_as_float(__builtin_amdgcn_ds_bpermute(srcLane << 2, __float_as_int(v)));
}

// Build the per-lane A-matrix fragments for all 16 K-chunks.
// A is 16x4 (MxK): row 0 = Wk chunk, row 1 = Wq chunk, rows 2..15 = 0.
// Layout (f32 16x4): lanes 0-15 hold M=lane with K={0,1} in (a0,a1);
// lanes 16-31 hold M=lane-16 with K={2,3}.
__device__ __forceinline__ void load_wfrag(const float* __restrict__ W, int lane,
                                           float a0[16], float a1[16]) {
    const int  m    = lane & 15;
    const int  koff = (lane & 16) ? 2 : 0;
    const bool act  = (m < 2);
    const float* p  = W + (act ? m : 0) * EMB + koff;   // Wk = W[0:64], Wq = W[64:128]
#pragma unroll
    for (int c = 0; c < 16; ++c) {
        const float w0 = p[4 * c + 0];
        const float w1 = p[4 * c + 1];
        a0[c] = act ? w0 : 0.0f;
        a1[c] = act ? w1 : 0.0f;
    }
}

// 16-chunk V_WMMA_F32_16X16X4_F32 chain: D[m][n] = dot(Wsel_m, row_n).
// B (4x16) fragments come from the wave-private LDS tile: lane l<16 supplies
// row (l) elems {4c,4c+1}; lane l>=16 supplies row (l-16) elems {4c+2,4c+3}.
__device__ __forceinline__ v8f wmma_dot_chain(const float* buf, int lane,
                                              const float a0[16], const float a1[16]) {
    v8f acc = {};
    const int row  = lane & 15;
    const int koff = (lane & 16) ? 2 : 0;
    const float* p = buf + row * ROWSTRIDE + koff;
#pragma unroll
    for (int c = 0; c < 16; ++c) {
        v2f b = *(const v2f*)(p + 4 * c);     // ds_load_b64, conflict-free (stride 68)
        v2f a; a.x = a0[c]; a.y = a1[c];
        acc = __builtin_amdgcn_wmma_f32_16x16x4_f32(false, a, false, b,
                                                    (short)0, acc, false, false);
    }
    return acc;
}

__device__ __forceinline__ float4 wsum3(float a, const float4& x,
                                        float b, const float4& y,
                                        float c, const float4& z) {
    float4 r;
    r.x = a * x.x + b * y.x + c * z.x;
    r.y = a * x.y + b * y.y + c * z.y;
    r.z = a * x.z + b * y.z + c * z.z;
    r.w = a * x.w + b * y.w + c * z.w;
    return r;
}

__global__ __launch_bounds__(BLOCK_THREADS)
void MGA_50766513439346_attn2(const float* __restrict__ buy,
                              const float* __restrict__ view,
                              const float* __restrict__ cart,
                              const float* __restrict__ view_buy,
                              const float* __restrict__ cart_buy,
                              const float* __restrict__ buy_buy,
                              const float* __restrict__ W0, const float* __restrict__ b0,
                              const float* __restrict__ W1, const float* __restrict__ b1,
                              float* __restrict__ out, int nRows, int nTiles) {
    __shared__ __align__(16) float lds[WAVES_PER_BLOCK * TILE_ROWS * ROWSTRIDE];

    const int lane = threadIdx.x & 31;
    const int wv   = threadIdx.x >> 5;
    const int tile = blockIdx.x * WAVES_PER_BLOCK + wv;
    if (tile >= nTiles) return;                 // wave-uniform: EXEC stays all-ones below
    float* buf = lds + wv * (TILE_ROWS * ROWSTRIDE);

    const int  sub     = lane >> 4;             // 0: even row of pair, 1: odd row
    const int  col     = (lane & 15) * 4;       // this lane's 4-float column slice
    const long rowBase = (long)tile * TILE_ROWS;

    // Coalesced global_load_b128 of one 16x64 tile into regs + LDS staging copy.
    auto load_tile = [&](const float* __restrict__ X, float4* regs) {
#pragma unroll
        for (int j = 0; j < 8; ++j) {
            long r = rowBase + 2 * j + sub;
            r = (r < nRows) ? r : (nRows - 1);  // clamp (select, no divergence)
            const float4 v = *(const float4*)(X + r * EMB + col);
            regs[j] = v;
            *(float4*)(buf + (2 * j + sub) * ROWSTRIDE + col) = v;
        }
    };

    float  a0[16], a1[16];
    float4 Xv[8], Xc[8], Xb[8];

    // ---------------- Stage 1: key = buy, Q = [view, cart, buy] ----------------
    load_wfrag(W0, lane, a0, a1);
    load_tile(view, Xv);  const v8f Dv = wmma_dot_chain(buf, lane, a0, a1);
    load_tile(cart, Xc);  const v8f Dc = wmma_dot_chain(buf, lane, a0, a1);
    load_tile(buy,  Xb);  const v8f Db = wmma_dot_chain(buf, lane, a0, a1);

    // Pull stage-2 operands toward GL2 while we do stage-1 math (global_prefetch_b8).
#pragma unroll
    for (int j = 0; j < 8; ++j) {
        long r = rowBase + 2 * j + sub;
        r = (r < nRows) ? r : (nRows - 1);
        __builtin_prefetch(view_buy + r * EMB + col, 0, 1);
        __builtin_prefetch(cart_buy + r * EMB + col, 0, 1);
        __builtin_prefetch(buy_buy  + r * EMB + col, 0, 1);
    }

    // lanes 0..15 hold row n's scalars: D[0] = Wk·row (key-dot), D[1] = Wq·row
    {
        const float bias = b0[0];
        const float kd   = Db[0];               // Wk0·buy_n
        const float s_v  = kd + Dv[1] + bias;
        const float s_c  = kd + Dc[1] + bias;
        const float s_b  = kd + Db[1] + bias;
        const float mx   = fmaxf(s_v, fmaxf(s_c, s_b));
        const float ev = __expf(s_v - mx), ec = __expf(s_c - mx), eb = __expf(s_b - mx);
        const float inv = 1.0f / (ev + ec + eb);
        const float av = ev * inv, ac = ec * inv, ab = eb * inv;

        // key1 = av*view + ac*cart + ab*buy  ->  LDS only (never touches HBM)
#pragma unroll
        for (int j = 0; j < 8; ++j) {
            const int r = 2 * j + sub;
            const float4 k = wsum3(lane_bcast(av, r), Xv[j],
                                   lane_bcast(ac, r), Xc[j],
                                   lane_bcast(ab, r), Xb[j]);
            *(float4*)(buf + r * ROWSTRIDE + col) = k;
        }
    }

    // ---------------- Stage 2: key = key1, Q = [view_buy, cart_buy, buy_buy] ----
    load_wfrag(W1, lane, a0, a1);
    const v8f Dk = wmma_dot_chain(buf, lane, a0, a1);          // Wk1·key1 at D[0]
    load_tile(view_buy, Xv);  const v8f D2v = wmma_dot_chain(buf, lane, a0, a1);
    load_tile(cart_buy, Xc);  const v8f D2c = wmma_dot_chain(buf, lane, a0, a1);
    load_tile(buy_buy,  Xb);  const v8f D2b = wmma_dot_chain(buf, lane, a0, a1);

    const float bias1 = b1[0];
    const float kd1   = Dk[0];
    const float t_v   = kd1 + D2v[1] + bias1;
    const float t_c   = kd1 + D2c[1] + bias1;
    const float t_b   = kd1 + D2b[1] + bias1;
    const float mx2   = fmaxf(t_v, fmaxf(t_c, t_b));
    const float e2v = __expf(t_v - mx2), e2c = __expf(t_c - mx2), e2b = __expf(t_b - mx2);
    const float inv2 = 1.0f / (e2v + e2c + e2b);
    const float av2 = e2v * inv2, ac2 = e2c * inv2, ab2 = e2b * inv2;

#pragma unroll
    for (int j = 0; j < 8; ++j) {
        const int r = 2 * j + sub;
        const float4 o = wsum3(lane_bcast(av2, r), Xv[j],
                               lane_bcast(ac2, r), Xc[j],
                               lane_bcast(ab2, r), Xb[j]);
        const long rg = rowBase + r;
        if (rg < nRows)                                   // divergence only after all WMMA
            *(float4*)(out + rg * EMB + col) = o;         // coalesced global_store_b128
    }
}

extern "C" void kernel_launch(void* const* d_in, const int* in_sizes, int n_in,
                              void* d_out, int out_size, void* d_ws, size_t ws_size,
                              hipStream_t stream) {
    const float* buy      = (const float*)d_in[0];
    const float* view     = (const float*)d_in[1];
    const float* cart     = (const float*)d_in[2];
    const float* view_buy = (const float*)d_in[3];
    const float* cart_buy = (const float*)d_in[4];
    const float* buy_buy  = (const float*)d_in[5];
    const float* W0 = (const float*)d_in[6];
    const float* b0 = (const float*)d_in[7];
    const float* W1 = (const float*)d_in[8];
    const float* b1 = (const float*)d_in[9];
    float* out = (float*)d_out;

    const int nRows  = in_sizes[0] / EMB;                       // 500000
    const int nTiles = (nRows + TILE_ROWS - 1) / TILE_ROWS;     // 31250 (exact fit)
    const int nBlk   = (nTiles + WAVES_PER_BLOCK - 1) / WAVES_PER_BLOCK;

    MGA_50766513439346_attn2<<<nBlk, BLOCK_THREADS, 0, stream>>>(
        buy, view, cart, view_buy, cart_buy, buy_buy, W0, b0, W1, b1,
        out, nRows, nTiles);
}